// MyGraphConv_86749749444624
// MI455X (gfx1250) — compile-verified
//
#include <hip/hip_runtime.h>

// ---------------------------------------------------------------------------
// GraphSAGE-style conv for MI455X (gfx1250, wave32).
//   agg[dst] += h[src]  (E=600000 edges, D=128)   -> L2-resident atomics
//   h_neigh = agg / max(deg,1)                     -> fused in-place scale
//   out = relu(concat(h_dst, h_neigh) @ W^T + b)   -> V_WMMA_F32_16X16X4_F32
// ---------------------------------------------------------------------------

typedef __attribute__((ext_vector_type(2))) float v2f;
typedef __attribute__((ext_vector_type(8))) float v8f;

#define D_FEAT 128
#define K_TOT  256
#define N_OUT  128

// ---- zero the workspace (agg + deg) ---------------------------------------
__global__ __launch_bounds__(256) void gc_zero(float* __restrict__ p, int n) {
    int i = blockIdx.x * 256 + threadIdx.x;
    if (i < n) p[i] = 0.0f;
}

// ---- edge scatter: one wave32 per edge, 4 coalesced atomics per lane ------
__global__ __launch_bounds__(256) void gc_scatter(const float* __restrict__ h,
                                                  const int*  __restrict__ esrc,
                                                  const int*  __restrict__ edst,
                                                  float* __restrict__ agg,
                                                  float* __restrict__ deg,
                                                  int E) {
    const int e    = blockIdx.x * 8 + (threadIdx.x >> 5);   // 8 waves / block
    const int lane = threadIdx.x & 31;
    if (e >= E) return;
    const int src = esrc[e];
    const int dst = edst[e];
    const float* __restrict__ hrow = h   + (size_t)src * D_FEAT;
    float*       __restrict__ arow = agg + (size_t)dst * D_FEAT;
#pragma unroll
    for (int j = 0; j < 4; ++j) {
        const int idx = lane + j * 32;                      // coalesced 128B/op
        atomicAdd(&arow[idx], hrow[idx]);                   // global_atomic_add_f32
    }
    if (lane == 0) atomicAdd(&deg[dst], 1.0f);
}

// ---- mean: agg[m][*] *= 1/max(deg[m],1) (in place) ------------------------
__global__ __launch_bounds__(256) void gc_mean(float* __restrict__ agg,
                                               const float* __restrict__ deg,
                                               int n_elems) {
    int i = blockIdx.x * 256 + threadIdx.x;
    if (i >= n_elems) return;
    const int m = i >> 7;                                   // /128
    const float d = deg[m];
    agg[i] = agg[i] * (1.0f / fmaxf(d, 1.0f));
}

// ---- GEMM + bias + ReLU via V_WMMA_F32_16X16X4_F32 ------------------------
// Block = 256 threads = 8 waves. Block b owns output rows [16b, 16b+16);
// wave w owns output cols [16w, 16w+16). K=256 -> 64 accumulating WMMAs.
// A lane layout (16x4 f32): row = lane%16, K-pair base = 2*(lane>>4).
// B lane layout (4x16 f32): col = lane%16, same K-pair striping.
// C/D (16x16 f32, 8 VGPRs):  M = v + 8*(lane>>4), N = lane%16.
__global__ __launch_bounds__(256) void gc_gemm_relu(const float* __restrict__ h,
                                                    const float* __restrict__ aggs,
                                                    const float* __restrict__ W,
                                                    const float* __restrict__ bias,
                                                    float* __restrict__ out,
                                                    int n_dst) {
    const int wave = threadIdx.x >> 5;           // 0..7  -> N tile
    const int lane = threadIdx.x & 31;
    const int row  = lane & 15;                  // A: M index / B: N index
    const int hi   = lane >> 4;                  // K-pair selector (0/1)
    const int m0   = blockIdx.x * 16;
    const int n0   = wave * 16;

    // Clamp row reads in case n_dst is not a multiple of 16 (it is: 20000).
    const int mr = min(m0 + row, n_dst - 1);

    const float* __restrict__ Ah = h    + (size_t)mr * D_FEAT        + 2 * hi;
    const float* __restrict__ An = aggs + (size_t)mr * D_FEAT        + 2 * hi;
    const float* __restrict__ Bw = W    + (size_t)(n0 + row) * K_TOT + 2 * hi;

    // Seed accumulator with the bias (C-matrix of first WMMA).
    const float bn = bias[n0 + row];
    v8f acc;
#pragma unroll
    for (int v = 0; v < 8; ++v) acc[v] = bn;

    // K = 0..127 : self features h_dst
#pragma unroll 8
    for (int k0 = 0; k0 < D_FEAT; k0 += 4) {
        v2f a = *(const v2f*)(Ah + k0);
        v2f b = *(const v2f*)(Bw + k0);
        acc = __builtin_amdgcn_wmma_f32_16x16x4_f32(
            /*neg_a=*/false, a, /*neg_b=*/false, b,
            /*c_mod=*/(short)0, acc, /*reuse_a=*/false, /*reuse_b=*/false);
    }
    // K = 128..255 : mean-aggregated neighbor features
#pragma unroll 8
    for (int k0 = 0; k0 < D_FEAT; k0 += 4) {
        v2f a = *(const v2f*)(An + k0);
        v2f b = *(const v2f*)(Bw + D_FEAT + k0);
        acc = __builtin_amdgcn_wmma_f32_16x16x4_f32(
            false, a, false, b, (short)0, acc, false, false);
    }

    // ReLU + store: out[(m0 + v + 8*hi)][n0 + row]
#pragma unroll
    for (int v = 0; v < 8; ++v) {
        const int m = m0 + v + 8 * hi;
        if (m < n_dst)
            out[(size_t)m * N_OUT + n0 + row] = fmaxf(acc[v], 0.0f);
    }
}

// ---------------------------------------------------------------------------
extern "C" void kernel_launch(void* const* d_in, const int* in_sizes, int n_in,
                              void* d_out, int out_size, void* d_ws, size_t ws_size,
                              hipStream_t stream) {
    const float* h    = (const float*)d_in[0];   // (N_SRC, 128) f32
    const int*   esrc = (const int*)  d_in[1];   // (E,) int32
    const int*   edst = (const int*)  d_in[2];   // (E,) int32
    const float* W    = (const float*)d_in[3];   // (128, 256) f32
    const float* bias = (const float*)d_in[4];   // (128,) f32
    // d_in[5] = n_dst scalar (device); derive on host instead:
    const int E     = in_sizes[1];
    const int n_dst = out_size / N_OUT;          // 20000

    float* agg = (float*)d_ws;                   // n_dst * 128 floats
    float* deg = agg + (size_t)n_dst * D_FEAT;   // n_dst floats
    float* out = (float*)d_out;

    const int zero_n = n_dst * D_FEAT + n_dst;
    gc_zero<<<(zero_n + 255) / 256, 256, 0, stream>>>(agg, zero_n);

    gc_scatter<<<(E + 7) / 8, 256, 0, stream>>>(h, esrc, edst, agg, deg, E);

    const int n_elems = n_dst * D_FEAT;
    gc_mean<<<(n_elems + 255) / 256, 256, 0, stream>>>(agg, deg, n_elems);

    gc_gemm_relu<<<(n_dst + 15) / 16, 256, 0, stream>>>(h, agg, W, bias, out, n_dst);
}